// ARCS_loss_23622320128391
// MI455X (gfx1250) — compile-verified
//
#include <hip/hip_runtime.h>
#include <cstdint>
#include <cstddef>

#define NUM_CLASS 19
#define N_PIX     262144
#define D_FEAT    256
#define CPAD      32          // classes padded to 2 WMMA N-tiles
#define TILES     (N_PIX / 16)

// ---- workspace layout (float indices) ----
#define SUMC_OFF    0          // 19*256 used
#define DENOM_OFF   8192       // 32
#define SUMWS_OFF   8224       // 1
#define TOTENT_OFF  8225       // 1
#define CENT_OFF    8256       // 32*256 cent_safe (padded, rows 19..31 = 0)
#define SEEN_OFF    16448      // 32
#define ZERO_FLOATS 8256       // region that must start at 0

#define WPB      4             // waves per block in k_entropy
#define ASTRIDE  36            // padded A-chunk row stride (floats): banks 36r mod 64 distinct

typedef float v2f __attribute__((ext_vector_type(2)));
typedef float v8f __attribute__((ext_vector_type(8)));

// ------------------------------------------------------------------
// CDNA5 async-to-LDS helpers (ASYNCcnt-tracked DMA, CDNA5 ISA)
// ------------------------------------------------------------------
__device__ __forceinline__ void async_b128(uint32_t lds_off, const float* g) {
    asm volatile("global_load_async_to_lds_b128 %0, %1, off"
                 :: "v"(lds_off), "v"((unsigned long long)(uintptr_t)g)
                 : "memory");
}
__device__ __forceinline__ void wait_async0() {
    asm volatile("s_wait_asynccnt 0x0" ::: "memory");
}
__device__ __forceinline__ void wait_async4() {
    asm volatile("s_wait_asynccnt 0x4" ::: "memory");
}

// ------------------------------------------------------------------
// Kernel 0: zero the accumulator region of the workspace
// ------------------------------------------------------------------
__global__ void k_zero(float* __restrict__ ws) {
    int i = blockIdx.x * blockDim.x + threadIdx.x;
    int stride = gridDim.x * blockDim.x;
    for (; i < ZERO_FLOATS; i += stride) ws[i] = 0.0f;
}

// ------------------------------------------------------------------
// Kernel 1: per-class weighted sums (both domains) via LDS privatization
//   blockDim.x must be 256 (== D_FEAT): thread t owns feature dim t.
// ------------------------------------------------------------------
__global__ void __launch_bounds__(256)
k_accum(const float* __restrict__ sfeat, const float* __restrict__ tfeat,
        const float* __restrict__ conf,  const int* __restrict__ sarg,
        const int* __restrict__ targ,    const unsigned char* __restrict__ smask,
        float* __restrict__ ws) {
    __shared__ float lsum[NUM_CLASS * D_FEAT];
    __shared__ float lden[NUM_CLASS];
    __shared__ float lws;
    const int tid = threadIdx.x;

    for (int i = tid; i < NUM_CLASS * D_FEAT; i += blockDim.x) lsum[i] = 0.0f;
    if (tid < NUM_CLASS) lden[tid] = 0.0f;
    if (tid == 0) lws = 0.0f;
    __syncthreads();

    for (int p = blockIdx.x; p < N_PIX; p += gridDim.x) {
        const int   cs  = sarg[p];
        const int   ct  = targ[p];
        const float wt  = 1.0f - conf[p];
        const bool  msk = smask[p] != 0;
        if (msk) {                       // skip load entirely when weight == 0
            float v = sfeat[(size_t)p * D_FEAT + tid];
            atomicAdd(&lsum[cs * D_FEAT + tid], v);       // ds_add_f32
        }
        float v2 = tfeat[(size_t)p * D_FEAT + tid];
        atomicAdd(&lsum[ct * D_FEAT + tid], v2 * wt);
        if (tid == 0) {
            if (msk) { atomicAdd(&lden[cs], 1.0f); atomicAdd(&lws, 1.0f); }
            atomicAdd(&lden[ct], wt);
        }
    }
    __syncthreads();

    for (int i = tid; i < NUM_CLASS * D_FEAT; i += blockDim.x)
        if (lsum[i] != 0.0f) atomicAdd(&ws[SUMC_OFF + i], lsum[i]);
    if (tid < NUM_CLASS && lden[tid] != 0.0f) atomicAdd(&ws[DENOM_OFF + tid], lden[tid]);
    if (tid == 0 && lws != 0.0f) atomicAdd(&ws[SUMWS_OFF], lws);
}

// ------------------------------------------------------------------
// Kernel 2: centroids (+inf for unseen) -> d_out; cent_safe + seen -> ws
// ------------------------------------------------------------------
__global__ void __launch_bounds__(256)
k_centroid(float* __restrict__ ws, float* __restrict__ out) {
    const int d = threadIdx.x;   // 0..255
    for (int c = 0; c < CPAD; ++c) {
        if (c < NUM_CLASS) {
            float den  = ws[DENOM_OFF + c];
            bool  sn   = den > 0.0f;
            float cent = sn ? ws[SUMC_OFF + c * D_FEAT + d] / fmaxf(den, 1e-12f)
                            : INFINITY;
            out[c * D_FEAT + d]           = cent;
            ws[CENT_OFF + c * D_FEAT + d] = sn ? cent : 0.0f;
        } else {
            ws[CENT_OFF + c * D_FEAT + d] = 0.0f;        // WMMA N-pad
        }
    }
    if (d < CPAD)
        ws[SEEN_OFF + d] = (d < NUM_CLASS && ws[DENOM_OFF + d] > 0.0f) ? 1.0f : 0.0f;
}

// ------------------------------------------------------------------
// Kernel 3: z = feat @ cent_safe.T via V_WMMA_F32_16X16X4_F32 with
//   double-buffered GLOBAL_LOAD_ASYNC_TO_LDS_B128 A-tile staging,
//   fused masked-entropy + weighted reduction.
//   128 threads = 4 waves; one wave per 16-pixel tile.
// ------------------------------------------------------------------
__global__ void __launch_bounds__(128)
k_entropy(const float* __restrict__ sfeat, const float* __restrict__ tfeat,
          const float* __restrict__ conf,  const unsigned char* __restrict__ smask,
          const float* __restrict__ ws,    float* __restrict__ tot) {
    __shared__ float centLds[CPAD][D_FEAT + 4];     // +4 pad: banks = 4c, distinct
    __shared__ float seenLds[CPAD];
    __shared__ float zbuf[WPB][16][33];             // per-wave z tile, stride 33
    __shared__ float abuf[WPB][2][16][ASTRIDE];     // per-wave double-buffered A chunks

    const int tid = threadIdx.x;

    // ---- stage cent_safe (32x256) into padded LDS via async DMA ----
    #pragma unroll
    for (int it = 0; it < 16; ++it) {
        int e = tid * 4 + it * 512;                 // float index, 128 thr * 4
        int r = e >> 8, c = e & 255;
        async_b128((uint32_t)(uintptr_t)&centLds[r][c], ws + CENT_OFF + e);
    }
    if (tid < CPAD) seenLds[tid] = ws[SEEN_OFF + tid];
    wait_async0();
    __syncthreads();

    const int lane  = tid & 31;
    const int wv    = tid >> 5;
    const int laneM = lane & 15;
    const int hf    = lane >> 4;                    // half-wave -> K block {2hf,2hf+1}
    const int gwave  = (blockIdx.x * blockDim.x + tid) >> 5;
    const int nwaves = (gridDim.x * blockDim.x) >> 5;

    float entAcc = 0.0f;

    for (int t = gwave; t < 2 * TILES; t += nwaves) {
        const bool   isSrc = t < TILES;
        const float* feat  = isSrc ? sfeat : tfeat;
        const int    row0  = (isSrc ? t : t - TILES) << 4;
        const float* tbase = feat + (size_t)row0 * D_FEAT;

        __builtin_prefetch(tbase + lane * 128, 0, 1);   // global_prefetch_b8

        // ---- issue async chunk kc into buffer b: 16 rows x 32 floats ----
        auto issueChunk = [&](int kc, int b) {
            #pragma unroll
            for (int i = 0; i < 4; ++i) {
                int e = i * 128 + lane * 4;             // float idx in 16x32 chunk
                int r = e >> 5;                         // 4 rows per instruction
                int c = e & 31;                         // 128B contiguous per row
                async_b128((uint32_t)(uintptr_t)&abuf[wv][b][r][c],
                           tbase + (size_t)r * D_FEAT + kc * 32 + c);
            }
        };

        issueChunk(0, 0);
        v8f acc0 = 0.0f;
        v8f acc1 = 0.0f;
        int bsel = 0;

        for (int kc = 0; kc < 8; ++kc) {
            if (kc < 7) { issueChunk(kc + 1, bsel ^ 1); wait_async4(); }
            else        { wait_async0(); }
            #pragma unroll
            for (int j = 0; j < 8; ++j) {
                v2f a  = *(const v2f*)&abuf[wv][bsel][laneM][j * 4 + 2 * hf];
                v2f b0 = *(const v2f*)&centLds[laneM]     [kc * 32 + j * 4 + 2 * hf];
                v2f b1 = *(const v2f*)&centLds[laneM + 16][kc * 32 + j * 4 + 2 * hf];
                acc0 = __builtin_amdgcn_wmma_f32_16x16x4_f32(
                           false, a, false, b0, (short)0, acc0, false, false);
                acc1 = __builtin_amdgcn_wmma_f32_16x16x4_f32(
                           false, a, false, b1, (short)0, acc1, false, false);
            }
            bsel ^= 1;
        }

        // D layout: acc[v] -> row hf*8+v, col laneM (+16 for acc1)
        #pragma unroll
        for (int v = 0; v < 8; ++v) {
            zbuf[wv][hf * 8 + v][laneM]      = acc0[v];
            zbuf[wv][hf * 8 + v][16 + laneM] = acc1[v];
        }
        asm volatile("s_wait_dscnt 0x0" ::: "memory");
        __builtin_amdgcn_wave_barrier();

        if (lane < 16) {                          // one lane per pixel row
            const int   p = row0 + lane;
            const float w = isSrc ? (smask[p] ? 1.0f : 0.0f) : (1.0f - conf[p]);
            if (w != 0.0f) {
                float m = -INFINITY;
                #pragma unroll
                for (int c = 0; c < NUM_CLASS; ++c)
                    if (seenLds[c] != 0.0f) m = fmaxf(m, zbuf[wv][lane][c]);
                float se = 0.0f;
                #pragma unroll
                for (int c = 0; c < NUM_CLASS; ++c)
                    if (seenLds[c] != 0.0f) se += __expf(zbuf[wv][lane][c] - m);
                const float lse = __logf(se);
                float ent = 0.0f;
                #pragma unroll
                for (int c = 0; c < NUM_CLASS; ++c)
                    if (seenLds[c] != 0.0f) {
                        float u = zbuf[wv][lane][c] - m - lse;
                        ent += __expf(u) * u;     // p * log p
                    }
                entAcc += w * ent;
            }
        }
        __builtin_amdgcn_wave_barrier();          // keep zbuf reuse ordered
    }

    // reduce across the wave (lanes >=16 hold 0)
    #pragma unroll
    for (int off = 16; off > 0; off >>= 1)
        entAcc += __shfl_xor(entAcc, off, 32);
    if (lane == 0) atomicAdd(tot, entAcc);        // global_atomic_add_f32
}

// ------------------------------------------------------------------
// Kernel 4: loss = -total / (sum(w_s) + N)
// ------------------------------------------------------------------
__global__ void k_final(const float* __restrict__ ws, float* __restrict__ out) {
    if (threadIdx.x == 0 && blockIdx.x == 0) {
        float total = ws[TOTENT_OFF];
        float n     = ws[SUMWS_OFF] + (float)N_PIX;
        out[NUM_CLASS * D_FEAT] = -total / n;
    }
}

// ------------------------------------------------------------------
extern "C" void kernel_launch(void* const* d_in, const int* in_sizes, int n_in,
                              void* d_out, int out_size, void* d_ws, size_t ws_size,
                              hipStream_t stream) {
    const float*         sfeat = (const float*)d_in[0];
    const float*         tfeat = (const float*)d_in[1];
    const float*         conf  = (const float*)d_in[2];
    const int*           sarg  = (const int*)d_in[3];
    const int*           targ  = (const int*)d_in[4];
    const unsigned char* smask = (const unsigned char*)d_in[5];
    float* out = (float*)d_out;
    float* ws  = (float*)d_ws;

    k_zero<<<64, 256, 0, stream>>>(ws);
    k_accum<<<2048, 256, 0, stream>>>(sfeat, tfeat, conf, sarg, targ, smask, ws);
    k_centroid<<<1, 256, 0, stream>>>(ws, out);
    k_entropy<<<2048, 128, 0, stream>>>(sfeat, tfeat, conf, smask, ws,
                                        ws + TOTENT_OFF);
    k_final<<<1, 64, 0, stream>>>(ws, out);
}